// MyModel_61933428415564
// MI455X (gfx1250) — compile-verified
//
#include <hip/hip_runtime.h>

// Overlap-add fold: out[o, s*8+j] = in[o, s, j] + in[o, s-1, 8+j]
// Shapes: in = (8,256,2000,16) f32, out = (8,256,16008) f32.
// Pure HBM-streaming kernel: 262MB in + 131MB out, each byte touched once.
// => non-temporal b128 loads/stores + prefetch; no WMMA (AI ~= 0.08 FLOP/B).

typedef __attribute__((ext_vector_type(4))) float v4f;

constexpr int    OUTER     = 8 * 256;              // 2048 independent rows
constexpr int    FRAMES    = 2000;
constexpr int    FRAME_LEN = 16;
constexpr int    STEP      = 8;
constexpr int    OUT_SUB   = FRAMES + 1;           // 2001 output subframes/row
constexpr size_t IN_ROW    = (size_t)FRAMES * FRAME_LEN;   // 32000 floats
constexpr size_t OUT_ROW   = (size_t)OUT_SUB * STEP;       // 16008 floats
constexpr int    PF_AHEAD  = 64;                   // prefetch 64 frames (4KB) ahead

__global__ __launch_bounds__(256)
void fold_overlap_add(const float* __restrict__ in, float* __restrict__ out) {
    const int s = blockIdx.x * blockDim.x + threadIdx.x;   // output subframe id
    const int o = blockIdx.y;                              // outer row id
    if (s >= OUT_SUB) return;

    const v4f* __restrict__ inrow  = (const v4f*)(in  + (size_t)o * IN_ROW);
    v4f*       __restrict__ outrow = (v4f*)      (out + (size_t)o * OUT_ROW);

    // Input bytes needed by this thread are the contiguous 64B window
    // [s*64 - 32, s*64 + 32): float4 indices 4s-2, 4s-1, 4s, 4s+1.
    const size_t p = 4 * (size_t)s;

    v4f a0 = {0.f, 0.f, 0.f, 0.f};   // frame s, first half
    v4f a1 = {0.f, 0.f, 0.f, 0.f};
    v4f b0 = {0.f, 0.f, 0.f, 0.f};   // frame s-1, second half
    v4f b1 = {0.f, 0.f, 0.f, 0.f};

    if (s < FRAMES) {
        a0 = __builtin_nontemporal_load(inrow + p);
        a1 = __builtin_nontemporal_load(inrow + p + 1);
        // Keep HBM request queues deep: gfx1250 global_prefetch_b8 into GL2.
        if (s + PF_AHEAD < FRAMES)
            __builtin_prefetch((const void*)(inrow + p + 4 * (size_t)PF_AHEAD), 0, 0);
    }
    if (s > 0) {
        b0 = __builtin_nontemporal_load(inrow + p - 2);
        b1 = __builtin_nontemporal_load(inrow + p - 1);
    }

    const v4f r0 = a0 + b0;
    const v4f r1 = a1 + b1;

    // Streaming store, never re-read: TH=NT.
    __builtin_nontemporal_store(r0, outrow + 2 * (size_t)s);
    __builtin_nontemporal_store(r1, outrow + 2 * (size_t)s + 1);
}

extern "C" void kernel_launch(void* const* d_in, const int* in_sizes, int n_in,
                              void* d_out, int out_size, void* d_ws, size_t ws_size,
                              hipStream_t stream) {
    (void)in_sizes; (void)n_in; (void)out_size; (void)d_ws; (void)ws_size;
    const float* in  = (const float*)d_in[0];
    float*       out = (float*)d_out;

    dim3 block(256);                                   // 8 wave32 waves / block
    dim3 grid((OUT_SUB + block.x - 1) / block.x,       // 8 blocks cover 2001 subframes
              OUTER);                                  // 2048 rows
    fold_overlap_add<<<grid, block, 0, stream>>>(in, out);
}